// LengthRegulator_19825569038381
// MI455X (gfx1250) — compile-verified
//
#include <hip/hip_runtime.h>
#include <stdint.h>

// Problem constants (from reference): B=16, TMAX=512, D=512, MAX_DUR=8
#define B_     16
#define TMAX_  512
#define D_     512
#define TF_    4096          // TMAX * MAX_DUR
#define YS_ELEMS ((size_t)B_ * TF_ * D_)   // 33,554,432 floats

typedef __attribute__((ext_vector_type(4))) float v4f;   // nontemporal-friendly 16B vector

// ---------------------------------------------------------------------------
// Kernel A: per-batch inclusive scan of durations -> frame->token index table.
// frame2t[b*TF_ + f] = t if starts[t] <= f < ends[t], else -1.
// One block per batch, 512 threads (16 wave32s).
// ---------------------------------------------------------------------------
__global__ __launch_bounds__(TMAX_) void lr_build_index(const int* __restrict__ ds,
                                                        int* __restrict__ frame2t) {
  __shared__ int sc[TMAX_];
  const int b = blockIdx.x;
  const int t = threadIdx.x;
  const int d = ds[b * TMAX_ + t];
  sc[t] = d;
  __syncthreads();

  // Hillis-Steele inclusive scan over 512 elements (9 steps).
  for (int off = 1; off < TMAX_; off <<= 1) {
    const int add = (t >= off) ? sc[t - off] : 0;
    __syncthreads();
    sc[t] += add;              // only thread t writes sc[t]
    __syncthreads();
  }
  const int end   = sc[t];
  const int start = end - d;

  // Initialize whole row of the table to -1 (covers frames >= total).
  int* row = frame2t + (size_t)b * TF_;
  for (int f = t; f < TF_; f += TMAX_) row[f] = -1;
  __syncthreads();

  // Each token claims its (<= 7) frames.
  for (int f = start; f < end; ++f) row[f] = t;
}

// ---------------------------------------------------------------------------
// Kernel B: expand. 2 output frame-rows per block; 128 lanes per row, one
// 16B vector per lane. map row = one-hot(t) via non-temporal stores.
// ys row = copy of xs[b, t, :] routed global -> LDS -> global with the CDNA5
// async data-mover ops (ASYNCcnt path), or zeros for tail frames.
// ---------------------------------------------------------------------------
__global__ __launch_bounds__(256) void lr_expand(const float* __restrict__ xs,
                                                 const int* __restrict__ frame2t,
                                                 float* __restrict__ ys,
                                                 float* __restrict__ mp) {
  __shared__ float stage[2][D_];          // 4 KB staging, one row per half-block

  const int tid = threadIdx.x;
  const int r   = tid >> 7;               // row within block (0/1)
  const int i   = tid & 127;              // 16B slot within row
  const int row = blockIdx.x * 2 + r;     // flat (b, f) row id
  const int b   = row >> 12;              // TF_ == 4096 == 2^12
  const int t   = frame2t[row];           // token for this frame, or -1
  const int c0  = i * 4;                  // first column handled by this lane

  v4f* ysv = (v4f*)(ys + (size_t)row * D_) + i;
  v4f* mpv = (v4f*)(mp + (size_t)row * D_) + i;

  // --- map: one-hot row, streamed with non-temporal hint (write-once data) ---
  v4f m;
  m.x = (t == c0    ) ? 1.0f : 0.0f;
  m.y = (t == c0 + 1) ? 1.0f : 0.0f;
  m.z = (t == c0 + 2) ? 1.0f : 0.0f;
  m.w = (t == c0 + 3) ? 1.0f : 0.0f;
  __builtin_nontemporal_store(m, mpv);

  // --- ys: gather-copy of xs[b, t, :] (branch is wave-uniform: t is per-row) --
  if (t >= 0) {
    const float* src = xs + ((size_t)(b * TMAX_ + t)) * D_ + c0;
    const unsigned lds_off = (unsigned)(uintptr_t)&stage[r][c0];

    // Async DMA: 16B/lane global -> LDS (tracked by ASYNCcnt).
    asm volatile("global_load_async_to_lds_b128 %0, %1, off"
                 :: "v"(lds_off), "v"((const void*)src)
                 : "memory");
    // Drain ASYNCcnt so the LDS bytes are in place before the store reads them.
    asm volatile("s_wait_asynccnt 0" ::: "memory");
    // Async DMA: 16B/lane LDS -> global ys row. s_endpgm waits for completion.
    asm volatile("global_store_async_from_lds_b128 %0, %1, off"
                 :: "v"((void*)ysv), "v"(lds_off)
                 : "memory");
  } else {
    v4f z = (v4f)0.0f;
    __builtin_nontemporal_store(z, ysv);
  }
}

// ---------------------------------------------------------------------------
// Host-side launcher (graph-capture safe: only kernel launches on `stream`).
// d_in[0] = xs (B,TMAX,D) f32; d_in[1] = ds (B,TMAX) int32.
// d_out = [ys | map] concatenated, 2 * 33,554,432 floats.
// d_ws  = frame->token table, B*TF_ ints (256 KB).
// ---------------------------------------------------------------------------
extern "C" void kernel_launch(void* const* d_in, const int* in_sizes, int n_in,
                              void* d_out, int out_size, void* d_ws, size_t ws_size,
                              hipStream_t stream) {
  (void)in_sizes; (void)n_in; (void)out_size; (void)ws_size;

  const float* xs = (const float*)d_in[0];
  const int*   ds = (const int*)d_in[1];
  float* ys = (float*)d_out;
  float* mp = (float*)d_out + YS_ELEMS;
  int*   frame2t = (int*)d_ws;

  lr_build_index<<<B_, TMAX_, 0, stream>>>(ds, frame2t);
  lr_expand<<<(B_ * TF_) / 2, 256, 0, stream>>>(xs, frame2t, ys, mp);
}